// FFF_43499428774596
// MI455X (gfx1250) — compile-verified
//
#include <hip/hip_runtime.h>
#include <math.h>

// MI455X / gfx1250, wave32.
// FFF tree: depth 11 -> 12 levels, 768-wide dots, 4095 nodes, batch 32768.
// Stage 1: scores for 16 samples/wave via V_WMMA_F32_16X16X4_F32 (diagonal of
//          W1gathered(16x4K) x Xtile^T(4Kx16)), dual accumulator chains.
// Stage 2: per-sample gather-AXPY over L2-resident w2s rows, float4 coalesced.
// Both stages prefetch the (known) next gather targets via global_prefetch_b8.

typedef float v2f __attribute__((ext_vector_type(2)));
typedef float v8f __attribute__((ext_vector_type(8)));

#define DEPTH            11
#define LEVELS           (DEPTH + 1)   // 12
#define DIN              768
#define DOUT             768
#define BATCH            32768
#define WAVES_PER_WG     8
#define THREADS          (WAVES_PER_WG * 32)
#define SAMPLES_PER_WAVE 16

__global__ __launch_bounds__(THREADS)
void fff_wmma_kernel(const float* __restrict__ x,
                     const float* __restrict__ w1s,
                     const float* __restrict__ w2s,
                     float* __restrict__ out)
{
    // per-wave scratch: gelu(score) and node id for each (level, sample)
    __shared__ float sG[WAVES_PER_WG][LEVELS][SAMPLES_PER_WAVE];
    __shared__ int   sN[WAVES_PER_WG][LEVELS][SAMPLES_PER_WAVE];

    const int tid   = threadIdx.x;
    const int lane  = tid & 31;
    const int wave  = tid >> 5;
    const int tile  = blockIdx.x * WAVES_PER_WG + wave;  // 16-sample tile
    const int s0    = tile * SAMPLES_PER_WAVE;

    const int col   = lane & 15;   // sample slot (B:N) == node slot (A:M)
    const int khalf = lane >> 4;   // K = khalf*2 + vgpr  (per ISA f32 layouts)

    // B operand (x tile) pointers for this lane, fixed across levels.
    // Split K in two halves -> two independent WMMA accumulate chains.
    const float* xrowL = x + (size_t)(s0 + col) * DIN + khalf * 2;
    const float* xrowH = xrowL + (DIN / 2);

    // source lane holding the true diagonal of column (lane&15):
    //  col i < 8  -> lane i      (VGPR i)
    //  col i >= 8 -> lane 16+i   (VGPR i-8)
    const int diagSrc = (col < 8) ? col : (16 + col);

    int myNode = 0;  // identical in lanes l and l+16 by construction

    // ---------------- stage 1: tree walk via WMMA F32 16x16x4 ----------------
    for (int k = 0; k < LEVELS; ++k) {
        // Prefetch BOTH children rows (contiguous 6KB at w1s[2n+1]) so the
        // next level's gather hits WGP$ regardless of the branch outcome.
        if (k < LEVELS - 1) {
            const char* ch = (const char*)(w1s + (size_t)(2 * myNode + 1) * DIN);
            __builtin_prefetch(ch + lane * 128, 0, 3);          // bytes 0..4095
            __builtin_prefetch(ch + 3072 + lane * 128, 0, 3);   // bytes 3072..7167
        }

        const float* arowL = w1s + (size_t)myNode * DIN + khalf * 2;
        const float* arowH = arowL + (DIN / 2);

        v8f c0 = {0.f, 0.f, 0.f, 0.f, 0.f, 0.f, 0.f, 0.f};
        v8f c1 = {0.f, 0.f, 0.f, 0.f, 0.f, 0.f, 0.f, 0.f};
#pragma unroll 2
        for (int kk = 0; kk < DIN / 8; ++kk) {        // 96 iters, 2 chains
            v2f aL = *(const v2f*)(arowL + kk * 4);   // A: gathered w1 rows
            v2f bL = *(const v2f*)(xrowL + kk * 4);   // B: x rows (transposed)
            v2f aH = *(const v2f*)(arowH + kk * 4);
            v2f bH = *(const v2f*)(xrowH + kk * 4);
            c0 = __builtin_amdgcn_wmma_f32_16x16x4_f32(
                     false, aL, false, bL, (short)0, c0, false, false);
            c1 = __builtin_amdgcn_wmma_f32_16x16x4_f32(
                     false, aH, false, bH, (short)0, c1, false, false);
        }
        v8f c = c0 + c1;

        // pick c[lane&7]  (valid diagonal lives in lanes 0..7 and 24..31)
        float e01 = (lane & 1) ? c[1] : c[0];
        float e23 = (lane & 1) ? c[3] : c[2];
        float e45 = (lane & 1) ? c[5] : c[4];
        float e67 = (lane & 1) ? c[7] : c[6];
        float eA  = (lane & 2) ? e23 : e01;
        float eB  = (lane & 2) ? e67 : e45;
        float ev  = (lane & 4) ? eB : eA;
        // broadcast column (lane&15)'s diagonal to both half-lanes
        float score = __shfl(ev, diagSrc, 32);

        if (lane < 16) {
            sN[wave][k][lane] = myNode;
            // exact-erf GELU, matches jax.nn.gelu(approximate=False)
            sG[wave][k][lane] =
                0.5f * score * (1.0f + erff(score * 0.70710678118654752f));
        }
        myNode = 2 * myNode + 1 + (score >= 0.0f ? 1 : 0);
    }

    // ------------- stage 2: out[s,:] = sum_k gelu_k * w2s[node_k,:] ----------
    // All 12 node ids per sample are known; prefetch next sample's rows while
    // accumulating the current one. w2s (12.6 MB) is L2-resident.
    for (int s = 0; s < SAMPLES_PER_WAVE; ++s) {
        if (s + 1 < SAMPLES_PER_WAVE) {
#pragma unroll
            for (int k = 0; k < LEVELS; ++k) {
                int ndn = sN[wave][k][s + 1];
                __builtin_prefetch(
                    (const char*)(w2s + (size_t)ndn * DOUT) + lane * 128, 0, 3);
            }
        }

        float4 acc[6];
#pragma unroll
        for (int j = 0; j < 6; ++j) acc[j] = make_float4(0.f, 0.f, 0.f, 0.f);

#pragma unroll
        for (int k = 0; k < LEVELS; ++k) {
            float g  = sG[wave][k][s];
            int   nd = sN[wave][k][s];
            const float4* row = (const float4*)(w2s + (size_t)nd * DOUT);
#pragma unroll
            for (int j = 0; j < 6; ++j) {
                float4 v = row[j * 32 + lane];  // coalesced 512B per instr
                acc[j].x += g * v.x;
                acc[j].y += g * v.y;
                acc[j].z += g * v.z;
                acc[j].w += g * v.w;
            }
        }

        float4* orow = (float4*)(out + (size_t)(s0 + s) * DOUT);
#pragma unroll
        for (int j = 0; j < 6; ++j) orow[j * 32 + lane] = acc[j];
    }
}

extern "C" void kernel_launch(void* const* d_in, const int* in_sizes, int n_in,
                              void* d_out, int out_size, void* d_ws, size_t ws_size,
                              hipStream_t stream)
{
    const float* x   = (const float*)d_in[0];
    const float* w1s = (const float*)d_in[1];
    const float* w2s = (const float*)d_in[2];
    // d_in[3] = depth (scalar, = 11); kernel is specialized for DEPTH=11
    float* out = (float*)d_out;

    dim3 grid(BATCH / (WAVES_PER_WG * SAMPLES_PER_WAVE));  // 256 blocks
    dim3 block(THREADS);                                   // 256 thr = 8 waves
    fff_wmma_kernel<<<grid, block, 0, stream>>>(x, w1s, w2s, out);
}